// DirectMultiHorizonDecoder_45105746543142
// MI455X (gfx1250) — compile-verified
//
#include <hip/hip_runtime.h>
#include <math.h>

// ---------------------------------------------------------------------------
// DirectMultiHorizonDecoder for MI455X (gfx1250), fp32 WMMA (V_WMMA_F32_16X16X4_F32)
// ---------------------------------------------------------------------------

typedef float v2f __attribute__((ext_vector_type(2)));
typedef float v8f __attribute__((ext_vector_type(8)));

#define WMMA4(a, b, c) \
  __builtin_amdgcn_wmma_f32_16x16x4_f32(false, (a), false, (b), (short)0, (c), false, false)

namespace {
constexpr int kB = 16, kS = 168, kN = 200, kH = 128, kT = 24, kNH = 4, kDH = 32;
constexpr long MKV = (long)kB * kS * kN;  // 537600 encoder tokens
constexpr long MQ  = (long)kB * kT * kN;  // 76800 decoder tokens
}

// ---------------------------------------------------------------------------
// Generic WMMA fp32 GEMM:  C[bz] = A[bz] (MxK) * B[bz] (KxN) [+ bias] [+ C]
// amode 0: A read directly.
// amode 1: A[m][k] = stepq[t][k] + hcomp[b*kN+n][k]   (m -> (b,t,n), K==kH)
// amode 2: A[m][k] = A[b*kN+n][k]                     (broadcast over t)
// Block: 256 threads (8 waves), tile 32(M) x 64(N), K staged in 32-chunks.
// Each wave owns one 16x16 WMMA tile.
// ---------------------------------------------------------------------------
__global__ __launch_bounds__(256)
void gemm_wmma(const float* __restrict__ A, const float* __restrict__ Bm,
               const float* __restrict__ bias, float* __restrict__ C,
               int M, int N, int K, int amode, int accum,
               long strideA, long strideB, long strideC,
               const float* __restrict__ stepq, const float* __restrict__ hcomp) {
  __shared__ float As[32 * 36];
  __shared__ float Bs[32 * 68];

  const int tid = threadIdx.x;
  const int bz  = blockIdx.z;
  const float* Ab = A ? (A + (long)bz * strideA) : nullptr;
  const float* Bb = Bm + (long)bz * strideB;
  float*       Cb = C  + (long)bz * strideC;

  const int m0 = blockIdx.y * 32;
  const int n0 = blockIdx.x * 64;
  const int wave = tid >> 5, lane = tid & 31;
  const int hi = lane >> 4, lo = lane & 15;
  const int tm = (wave >> 2) * 16;  // 0 | 16
  const int tn = (wave & 3) * 16;   // 0..48

  v8f acc = {0.f, 0.f, 0.f, 0.f, 0.f, 0.f, 0.f, 0.f};

  for (int k0 = 0; k0 < K; k0 += 32) {
    // Stage A tile 32x32 (zero-padded)
    for (int idx = tid; idx < 32 * 32; idx += 256) {
      int r = idx >> 5, c = idx & 31;
      int gm = m0 + r, gk = k0 + c;
      float v = 0.f;
      if (gm < M && gk < K) {
        if (amode == 0) {
          v = Ab[(long)gm * K + gk];
        } else {
          int b   = gm / (kT * kN);
          int rem = gm - b * (kT * kN);
          int t   = rem / kN;
          int n   = rem - t * kN;
          if (amode == 1)
            v = stepq[(long)t * kH + gk] + hcomp[((long)b * kN + n) * kH + gk];
          else
            v = Ab[((long)b * kN + n) * K + gk];
        }
      }
      As[r * 36 + c] = v;
    }
    // Stage B tile 32x64 (zero-padded)
    for (int idx = tid; idx < 32 * 64; idx += 256) {
      int r = idx >> 6, c = idx & 63;
      int gk = k0 + r, gn = n0 + c;
      Bs[r * 68 + c] = (gk < K && gn < N) ? Bb[(long)gk * N + gn] : 0.f;
    }
    __syncthreads();

#pragma unroll
    for (int kk = 0; kk < 32; kk += 4) {
      v2f a, b;
      a.x = As[(tm + lo) * 36 + kk + 2 * hi];
      a.y = As[(tm + lo) * 36 + kk + 2 * hi + 1];
      b.x = Bs[(kk + 2 * hi) * 68 + tn + lo];
      b.y = Bs[(kk + 2 * hi + 1) * 68 + tn + lo];
      acc = WMMA4(a, b, acc);
    }
    __syncthreads();
  }

#pragma unroll
  for (int r = 0; r < 8; ++r) {
    int gm = m0 + tm + r + 8 * hi;
    int gn = n0 + tn + lo;
    if (gm < M && gn < N) {
      float v = acc[r];
      if (bias) v += bias[gn];
      if (accum) v += Cb[(long)gm * N + gn];
      Cb[(long)gm * N + gn] = v;
    }
  }
}

// ---------------------------------------------------------------------------
// Fused per-(b, n, head) temporal cross-attention.
// scores = q(24x32) @ k^T(32x168) / sqrt(32); softmax over s; ctx = attn @ v.
// Writes attn in (B,T,NH,N,S) layout and ctx in (B,T,N,H) layout.
// ---------------------------------------------------------------------------
__global__ __launch_bounds__(256)
void attn_kernel(const float* __restrict__ qb, const float* __restrict__ kb,
                 const float* __restrict__ vb, float* __restrict__ attn_out,
                 float* __restrict__ ctx) {
  __shared__ float Ql[32 * 33];    // q rows 24..31 zero-padded
  __shared__ float KVl[176 * 33];  // K first, reused for V (rows 168..175 zero)
  __shared__ float Sl[32 * 177];   // scores / attn

  const int wg = blockIdx.x;
  const int h  = wg % kNH;
  const int n  = (wg / kNH) % kN;
  const int b  = wg / (kNH * kN);
  const int tid = threadIdx.x;
  const int wave = tid >> 5, lane = tid & 31;
  const int hi = lane >> 4, lo = lane & 15;

  // Load K (zero pad rows >= 168) and Q (zero pad rows >= 24)
  for (int idx = tid; idx < 176 * 32; idx += 256) {
    int s = idx >> 5, d = idx & 31;
    float v = 0.f;
    if (s < kS) v = kb[(((long)b * kS + s) * kN + n) * kH + h * kDH + d];
    KVl[s * 33 + d] = v;
  }
  for (int idx = tid; idx < 32 * 32; idx += 256) {
    int t = idx >> 5, d = idx & 31;
    float v = 0.f;
    if (t < kT) v = qb[(((long)b * kT + t) * kN + n) * kH + h * kDH + d];
    Ql[t * 33 + d] = v;
  }
  __syncthreads();

  const float scale = 0.17677669529663687f;  // 1/sqrt(32)

  // Scores: 2 (M) x 11 (N) tiles of 16x16, K = 32
  for (int job = wave; job < 22; job += 8) {
    int tm = (job / 11) * 16;
    int tn = (job % 11) * 16;
    v8f acc = {0.f, 0.f, 0.f, 0.f, 0.f, 0.f, 0.f, 0.f};
#pragma unroll
    for (int kk = 0; kk < 32; kk += 4) {
      v2f a, bfr;
      a.x = Ql[(tm + lo) * 33 + kk + 2 * hi];
      a.y = Ql[(tm + lo) * 33 + kk + 2 * hi + 1];
      // B[kd][s] = K[s][kd]
      bfr.x = KVl[(tn + lo) * 33 + kk + 2 * hi];
      bfr.y = KVl[(tn + lo) * 33 + kk + 2 * hi + 1];
      acc = WMMA4(a, bfr, acc);
    }
#pragma unroll
    for (int r = 0; r < 8; ++r)
      Sl[(tm + r + 8 * hi) * 177 + tn + lo] = acc[r] * scale;
  }
  __syncthreads();

  // Reload V into the K buffer (keeps static LDS < 64 KB)
  for (int idx = tid; idx < 176 * 32; idx += 256) {
    int s = idx >> 5, d = idx & 31;
    float v = 0.f;
    if (s < kS) v = vb[(((long)b * kS + s) * kN + n) * kH + h * kDH + d];
    KVl[s * 33 + d] = v;
  }

  // Softmax over s per row t (wave-per-row), write attn to global
  for (int t = wave; t < kT; t += 8) {
    float m = -1e30f;
    for (int s = lane; s < kS; s += 32) m = fmaxf(m, Sl[t * 177 + s]);
#pragma unroll
    for (int off = 16; off > 0; off >>= 1) m = fmaxf(m, __shfl_xor(m, off, 32));
    float sum = 0.f;
    for (int s = lane; s < kS; s += 32) {
      float e = expf(Sl[t * 177 + s] - m);
      Sl[t * 177 + s] = e;
      sum += e;
    }
#pragma unroll
    for (int off = 16; off > 0; off >>= 1) sum += __shfl_xor(sum, off, 32);
    float inv = 1.f / sum;
    for (int s = lane; s < kS; s += 32) {
      float a = Sl[t * 177 + s] * inv;
      Sl[t * 177 + s] = a;
      attn_out[((((long)b * kT + t) * kNH + h) * kN + n) * (long)kS + s] = a;
    }
  }
  __syncthreads();

  // ctx = attn(24x168) @ V(168x32): 2x2 tiles, K = 168
  for (int job = wave; job < 4; job += 8) {
    int tm = (job >> 1) * 16, tn = (job & 1) * 16;
    v8f acc = {0.f, 0.f, 0.f, 0.f, 0.f, 0.f, 0.f, 0.f};
    for (int kk = 0; kk < kS; kk += 4) {
      v2f a, bfr;
      a.x = Sl[(tm + lo) * 177 + kk + 2 * hi];
      a.y = Sl[(tm + lo) * 177 + kk + 2 * hi + 1];
      bfr.x = KVl[(kk + 2 * hi) * 33 + tn + lo];
      bfr.y = KVl[(kk + 2 * hi + 1) * 33 + tn + lo];
      acc = WMMA4(a, bfr, acc);
    }
#pragma unroll
    for (int r = 0; r < 8; ++r) {
      int t = tm + r + 8 * hi;
      if (t < kT)
        ctx[(((long)b * kT + t) * kN + n) * kH + h * kDH + tn + lo] = acc[r];
    }
  }
}

// ---------------------------------------------------------------------------
// LayerNorm over H=128, one wave per token (population variance, eps=1e-5)
// ---------------------------------------------------------------------------
__global__ __launch_bounds__(256)
void ln_kernel(const float* __restrict__ x, const float* __restrict__ g,
               const float* __restrict__ bta, float* __restrict__ y, int Mtok) {
  int tok = (int)((blockIdx.x * 256 + threadIdx.x) >> 5);
  int lane = threadIdx.x & 31;
  if (tok >= Mtok) return;
  const float* xp = x + (long)tok * kH;
  float v[4];
  float s = 0.f, s2 = 0.f;
#pragma unroll
  for (int c = 0; c < 4; ++c) {
    v[c] = xp[lane * 4 + c];
    s += v[c];
    s2 += v[c] * v[c];
  }
#pragma unroll
  for (int off = 16; off > 0; off >>= 1) {
    s  += __shfl_xor(s, off, 32);
    s2 += __shfl_xor(s2, off, 32);
  }
  float mean = s * (1.f / 128.f);
  float var  = s2 * (1.f / 128.f) - mean * mean;
  float inv  = rsqrtf(var + 1e-5f);
  float* yp = y + (long)tok * kH;
#pragma unroll
  for (int c = 0; c < 4; ++c) {
    int j = lane * 4 + c;
    yp[j] = (v[c] - mean) * inv * g[j] + bta[j];
  }
}

// ---------------------------------------------------------------------------
// GraphLSTM gate activation: gates (M x 4H) -> out (M x H)
// c broadcast over T; optional residual add.
// ---------------------------------------------------------------------------
__global__ __launch_bounds__(256)
void lstm_act(const float* __restrict__ gates, const float* __restrict__ cprev,
              const float* __restrict__ res, float* __restrict__ out, long total) {
  long idx = (long)blockIdx.x * blockDim.x + threadIdx.x;
  if (idx >= total) return;
  long tok = idx >> 7;
  int j = (int)(idx & 127);
  int b   = (int)(tok / (kT * kN));
  int rem = (int)(tok - (long)b * (kT * kN));
  int n   = rem % kN;
  const float* gp = gates + tok * 512;
  float gi = gp[j], gf = gp[128 + j], gg = gp[256 + j], go = gp[384 + j];
  float cp = cprev[((long)b * kN + n) * kH + j];
  float si = 1.f / (1.f + expf(-gi));
  float sf = 1.f / (1.f + expf(-gf));
  float so = 1.f / (1.f + expf(-go));
  float cn = sf * cp + si * tanhf(gg);
  float hv = so * tanhf(cn);
  if (res) hv += res[idx];
  out[idx] = hv;
}

// ---------------------------------------------------------------------------
// Final projection: preds = x @ out_W(128x1) + out_b, one wave per token
// ---------------------------------------------------------------------------
__global__ __launch_bounds__(256)
void outproj(const float* __restrict__ x, const float* __restrict__ w,
             const float* __restrict__ bb, float* __restrict__ preds, int Mtok) {
  int tok = (int)((blockIdx.x * 256 + threadIdx.x) >> 5);
  int lane = threadIdx.x & 31;
  if (tok >= Mtok) return;
  const float* xp = x + (long)tok * kH;
  float s = 0.f;
#pragma unroll
  for (int c = 0; c < 4; ++c) {
    int j = lane * 4 + c;
    s += xp[j] * w[j];
  }
#pragma unroll
  for (int off = 16; off > 0; off >>= 1) s += __shfl_xor(s, off, 32);
  if (lane == 0) preds[tok] = s + bb[0];
}

// ---------------------------------------------------------------------------
extern "C" void kernel_launch(void* const* d_in, const int* in_sizes, int n_in,
                              void* d_out, int out_size, void* d_ws, size_t ws_size,
                              hipStream_t stream) {
  (void)in_sizes; (void)n_in; (void)out_size; (void)ws_size;
  const float* enc   = (const float*)d_in[0];
  const float* h0    = (const float*)d_in[1];
  const float* c0    = (const float*)d_in[2];
  const float* h1    = (const float*)d_in[3];
  const float* c1    = (const float*)d_in[4];
  const float* adj   = (const float*)d_in[5];
  const float* stepq = (const float*)d_in[6];
  const float* Wq    = (const float*)d_in[7];
  const float* bq    = (const float*)d_in[8];
  const float* Wk    = (const float*)d_in[9];
  const float* bk    = (const float*)d_in[10];
  const float* Wv    = (const float*)d_in[11];
  const float* bv    = (const float*)d_in[12];
  const float* Wo    = (const float*)d_in[13];
  const float* bo    = (const float*)d_in[14];
  const float* ctxW  = (const float*)d_in[15];
  const float* ctxb  = (const float*)d_in[16];
  const float* Wx0   = (const float*)d_in[17];
  const float* Wh0   = (const float*)d_in[18];
  const float* b0    = (const float*)d_in[19];
  const float* lng0  = (const float*)d_in[20];
  const float* lnb0  = (const float*)d_in[21];
  const float* Wx1   = (const float*)d_in[22];
  const float* Wh1   = (const float*)d_in[23];
  const float* b1    = (const float*)d_in[24];
  const float* lng1  = (const float*)d_in[25];
  const float* lnb1  = (const float*)d_in[26];
  const float* outW  = (const float*)d_in[27];
  const float* outb  = (const float*)d_in[28];

  float* out = (float*)d_out;          // [0, MQ) preds, [MQ, ...) attn
  float* ws  = (float*)d_ws;

  // Workspace carve (floats). Post-attention buffers overlay dead K/V/Q.
  float* Kb   = ws;                        // 68,812,800
  float* Vb   = ws + 68812800L;            // 68,812,800
  float* Qb   = ws + 137625600L;           //  9,830,400
  float* Cx   = ws + 147456000L;           //  9,830,400 (raw attention ctx)
  float* Ctx1 = Qb;                        // ctx @ Wo (q dead after attention)
  float* Comb = Kb;                        // combined   (K dead)
  float* LNB  = Kb + 9830400L;             // layernorm scratch
  float* GT   = Kb + 19660800L;            // gates (39,321,600)
  float* X1   = Kb + 58982400L;            // layer-0 output
  float* X2   = Vb;                        // layer-1 output (V dead)
  float* AGG  = Vb + 9830400L;             // adj @ h (B,N,H)

  const dim3 blk(256);
  const dim3 gKV(2, (unsigned)((MKV + 31) / 32), 1);   // 537600 x 128
  const dim3 gQ(2, (unsigned)((MQ + 31) / 32), 1);     //  76800 x 128
  const dim3 gG(8, (unsigned)((MQ + 31) / 32), 1);     //  76800 x 512
  const dim3 gAdj(2, 7, kB);                           //    200 x 128 x16

  // 1-2. K/V projections over encoder tokens
  gemm_wmma<<<gKV, blk, 0, stream>>>(enc, Wk, bk, Kb, (int)MKV, 128, 128, 0, 0, 0, 0, 0, nullptr, nullptr);
  gemm_wmma<<<gKV, blk, 0, stream>>>(enc, Wv, bv, Vb, (int)MKV, 128, 128, 0, 0, 0, 0, 0, nullptr, nullptr);
  // 3. q = (step_q + h1) @ Wq + bq  (query composed on the fly)
  gemm_wmma<<<gQ, blk, 0, stream>>>(nullptr, Wq, bq, Qb, (int)MQ, 128, 128, 1, 0, 0, 0, 0, stepq, h1);
  // 4. fused attention: scores, softmax (-> attn output), ctx
  attn_kernel<<<dim3(kB * kN * kNH), blk, 0, stream>>>(Qb, Kb, Vb, out + MQ, Cx);
  // 5. ctx1 = ctx @ Wo + bo
  gemm_wmma<<<gQ, blk, 0, stream>>>(Cx, Wo, bo, Ctx1, (int)MQ, 128, 128, 0, 0, 0, 0, 0, nullptr, nullptr);
  // 6-7. combined = query @ ctxW[:128] + ctx1 @ ctxW[128:] + ctx_b
  gemm_wmma<<<gQ, blk, 0, stream>>>(nullptr, ctxW, ctxb, Comb, (int)MQ, 128, 128, 1, 0, 0, 0, 0, stepq, h1);
  gemm_wmma<<<gQ, blk, 0, stream>>>(Ctx1, ctxW + 128 * 128, nullptr, Comb, (int)MQ, 128, 128, 0, 1, 0, 0, 0, nullptr, nullptr);
  // 8. LN layer 0
  ln_kernel<<<dim3((unsigned)((MQ + 7) / 8)), blk, 0, stream>>>(Comb, lng0, lnb0, LNB, (int)MQ);
  // 9. agg0 = adj @ h0  (batched over B, shared adj)
  gemm_wmma<<<gAdj, blk, 0, stream>>>(adj, h0, nullptr, AGG, kN, 128, kN, 0, 0, 0, (long)kN * kH, (long)kN * kH, nullptr, nullptr);
  // 10-11. gates = ln @ Wx0 + b0 ; gates += agg0(bcast T) @ Wh0
  gemm_wmma<<<gG, blk, 0, stream>>>(LNB, Wx0, b0, GT, (int)MQ, 512, 128, 0, 0, 0, 0, 0, nullptr, nullptr);
  gemm_wmma<<<gG, blk, 0, stream>>>(AGG, Wh0, nullptr, GT, (int)MQ, 512, 128, 2, 1, 0, 0, 0, nullptr, nullptr);
  // 12. x1 = cell activation (no residual)
  lstm_act<<<dim3((unsigned)((MQ * kH) / 256)), blk, 0, stream>>>(GT, c0, nullptr, X1, MQ * kH);
  // 13. LN layer 1
  ln_kernel<<<dim3((unsigned)((MQ + 7) / 8)), blk, 0, stream>>>(X1, lng1, lnb1, LNB, (int)MQ);
  // 14. agg1 = adj @ h1
  gemm_wmma<<<gAdj, blk, 0, stream>>>(adj, h1, nullptr, AGG, kN, 128, kN, 0, 0, 0, (long)kN * kH, (long)kN * kH, nullptr, nullptr);
  // 15-16. gates = ln @ Wx1 + b1 ; gates += agg1(bcast T) @ Wh1
  gemm_wmma<<<gG, blk, 0, stream>>>(LNB, Wx1, b1, GT, (int)MQ, 512, 128, 0, 0, 0, 0, 0, nullptr, nullptr);
  gemm_wmma<<<gG, blk, 0, stream>>>(AGG, Wh1, nullptr, GT, (int)MQ, 512, 128, 2, 1, 0, 0, 0, nullptr, nullptr);
  // 17. x2 = cell activation + residual x1
  lstm_act<<<dim3((unsigned)((MQ * kH) / 256)), blk, 0, stream>>>(GT, c1, X1, X2, MQ * kH);
  // 18. preds = x2 @ out_W + out_b
  outproj<<<dim3((unsigned)((MQ + 7) / 8)), blk, 0, stream>>>(X2, outW, outb, out, (int)MQ);
}